// TopkGate_81784767250726
// MI455X (gfx1250) — compile-verified
//
#include <hip/hip_runtime.h>
#include <hip/hip_bf16.h>

typedef __attribute__((ext_vector_type(2))) float v2f;
typedef __attribute__((ext_vector_type(4))) float v4f;
typedef __attribute__((ext_vector_type(8))) float v8f;

// Problem sizes (fixed by setup_inputs in the reference)
#define B_SZ   8192
#define D_SZ   512
#define E_SZ   16
#define DIN_SZ 512
#define P_SZ   4

#define AVG_BLOCKS 64
#define AVG_ROWS   (B_SZ / AVG_BLOCKS)   // 128 rows per partial block

// ---------------------------------------------------------------------------
// Kernel 1: gate logits via full-fp32 WMMA (16x16x4), then per-row top-2,
// masked softmax, permutation mix, normalize. One wave32 per 16 rows.
// Writes g_perm [B, 16] to workspace.
// ---------------------------------------------------------------------------
__global__ __launch_bounds__(32) void gate_kernel(
    const float* __restrict__ x,     // [B, DIN]
    const float* __restrict__ gw,    // [E, DIN]  (gate_weights)
    const float* __restrict__ pwt,   // [P, E, E] (permutation_weights)
    const float* __restrict__ bias,  // [E]
    float* __restrict__ g_out)       // [B, E]
{
  __shared__ float lds_pw[E_SZ * E_SZ];      // mean over P of permutation_weights
  __shared__ float lds_logits[16 * 16];      // one 16x16 logits tile

  const int lane = threadIdx.x;              // 0..31 (wave32)
  const int tile = blockIdx.x;               // 16 rows per tile
  const int m    = lane & 15;                // row within tile for A, col N for B
  const int kh   = (lane >> 4) * 2;          // lanes 16-31 hold K+2,K+3

  // pw = mean over P (tiny: 256 values)
  for (int i = lane; i < E_SZ * E_SZ; i += 32) {
    lds_pw[i] = 0.25f * (pwt[i] + pwt[256 + i] + pwt[512 + i] + pwt[768 + i]);
  }

  // A: x rows of this tile.  B: gate_weights^T (logits = x @ gw^T).
  const float* xrow  = x  + (size_t)(tile * 16 + m) * DIN_SZ;
  const float* gwrow = gw + (size_t)m * DIN_SZ;   // N = lane&15 -> expert row of gw

  v8f c = {};  // f32 accumulator, RNE, full precision
  for (int k0 = 0; k0 < DIN_SZ; k0 += 4) {
    v2f a = *(const v2f*)(xrow  + k0 + kh);   // A[m, k0+kh], A[m, k0+kh+1]
    v2f b = *(const v2f*)(gwrow + k0 + kh);   // B[k0+kh, n], B[k0+kh+1, n]
    c = __builtin_amdgcn_wmma_f32_16x16x4_f32(
        /*neg_a=*/false, a, /*neg_b=*/false, b,
        /*c_mod=*/(short)0, c, /*reuse_a=*/false, /*reuse_b=*/false);
  }

  // Spill C tile to LDS: VGPR r holds M = r + 8*(lane>=16), N = lane&15
  {
    const int mhi = (lane >> 4) * 8;
    const int n   = lane & 15;
#pragma unroll
    for (int r = 0; r < 8; ++r)
      lds_logits[(mhi + r) * 16 + n] = c[r];
  }
  __syncthreads();

  // Per-row scalar tail: lanes 0..15 each own one row.
  if (lane < 16) {
    const int brow = tile * 16 + lane;
    float vals[16];
#pragma unroll
    for (int e = 0; e < 16; ++e) vals[e] = lds_logits[lane * 16 + e] + bias[e];

    // top-2 (descending, lowest index wins ties — matches jax.lax.top_k)
    int i1 = 0; float v1 = vals[0];
#pragma unroll
    for (int e = 1; e < 16; ++e) { if (vals[e] > v1) { v1 = vals[e]; i1 = e; } }
    int i2 = (i1 == 0) ? 1 : 0; float v2 = vals[i2];
#pragma unroll
    for (int e = 0; e < 16; ++e) {
      if (e != i1 && vals[e] > v2) { v2 = vals[e]; i2 = e; }
    }

    // faithful to tf.where(scattered == 0): an exactly-zero top-k logit is
    // masked to -inf too; all non-top-k are -inf.
    const bool k1 = (v1 != 0.0f);
    const bool k2 = (v2 != 0.0f);
    const float mx = k1 ? v1 : v2;                 // v1 >= v2 always
    const float e1 = k1 ? __expf(v1 - mx) : 0.0f;
    const float e2 = k2 ? __expf(v2 - mx) : 0.0f;
    const float inv = 1.0f / (e1 + e2);            // NaN if both masked (ref does too)
    const float g1 = e1 * inv, g2 = e2 * inv;

    // g_perm[j] = g1*pw[i1][j] + g2*pw[i2][j]; then row-normalize
    float gp[16]; float s = 0.0f;
#pragma unroll
    for (int j = 0; j < 16; ++j) {
      float t = g1 * lds_pw[i1 * 16 + j] + g2 * lds_pw[i2 * 16 + j];
      gp[j] = t; s += t;
    }
    const float invs = 1.0f / s;
    float* out = g_out + (size_t)brow * E_SZ;
#pragma unroll
    for (int j = 0; j < 16; ++j) out[j] = gp[j] * invs;
  }
}

// ---------------------------------------------------------------------------
// Kernel 2: y[b,d] = sum_e f[b,d,e] * g_perm[b,e].  Pure HBM bandwidth:
// 256 MB of f streamed once (> 192 MB L2, zero reuse) -> non-temporal b128
// loads; y written once -> non-temporal stores.  e is contiguous in memory.
// ---------------------------------------------------------------------------
__global__ __launch_bounds__(256) void y_kernel(
    const float* __restrict__ f,     // [B, D, E]
    const float* __restrict__ g,     // [B, E]
    float* __restrict__ y)           // [B, D]
{
  __shared__ float gs[E_SZ];
  const int b = blockIdx.x;
  const int t = threadIdx.x;
  if (t < E_SZ) gs[t] = g[(size_t)b * E_SZ + t];
  __syncthreads();

  float gv[16];
#pragma unroll
  for (int j = 0; j < 16; ++j) gv[j] = gs[j];

  const float* fb = f + (size_t)b * D_SZ * E_SZ;
  float* yb = y + (size_t)b * D_SZ;

  for (int d = t; d < D_SZ; d += 256) {
    const float* fr = fb + (size_t)d * E_SZ;      // 16 contiguous floats (64B)
    v4f f0 = __builtin_nontemporal_load((const v4f*)(fr + 0));
    v4f f1 = __builtin_nontemporal_load((const v4f*)(fr + 4));
    v4f f2 = __builtin_nontemporal_load((const v4f*)(fr + 8));
    v4f f3 = __builtin_nontemporal_load((const v4f*)(fr + 12));
    float acc = 0.0f;
    acc = fmaf(f0.x, gv[0],  acc); acc = fmaf(f0.y, gv[1],  acc);
    acc = fmaf(f0.z, gv[2],  acc); acc = fmaf(f0.w, gv[3],  acc);
    acc = fmaf(f1.x, gv[4],  acc); acc = fmaf(f1.y, gv[5],  acc);
    acc = fmaf(f1.z, gv[6],  acc); acc = fmaf(f1.w, gv[7],  acc);
    acc = fmaf(f2.x, gv[8],  acc); acc = fmaf(f2.y, gv[9],  acc);
    acc = fmaf(f2.z, gv[10], acc); acc = fmaf(f2.w, gv[11], acc);
    acc = fmaf(f3.x, gv[12], acc); acc = fmaf(f3.y, gv[13], acc);
    acc = fmaf(f3.z, gv[14], acc); acc = fmaf(f3.w, gv[15], acc);
    __builtin_nontemporal_store(acc, yb + d);
  }
}

// ---------------------------------------------------------------------------
// Kernel 3a: per-block partial sums over a fixed contiguous row range
// (deterministic, no atomics).  Lanes 0-15: soft; lanes 16-31: hard.
// ---------------------------------------------------------------------------
__global__ __launch_bounds__(32) void avg_partial_kernel(
    const float* __restrict__ g,     // [B, E]
    float* __restrict__ part)        // [AVG_BLOCKS, 32]
{
  const int t = threadIdx.x;
  const int j = t & 15;
  const bool hard = (t >= 16);
  const int b0 = blockIdx.x * AVG_ROWS;
  float s = 0.0f;
  for (int b = b0; b < b0 + AVG_ROWS; ++b) {
    const float v = g[(size_t)b * E_SZ + j];
    // hard: 1 - where(v < 1e-5, 1, 0) == !(v < 1e-5)  (NaN -> 1, same as ref)
    s += hard ? ((v < 1e-5f) ? 0.0f : 1.0f) : v;
  }
  part[blockIdx.x * 32 + t] = s;
}

// ---------------------------------------------------------------------------
// Kernel 3b: fixed-order final sum of the 64 partials -> soft/hard averages.
// ---------------------------------------------------------------------------
__global__ __launch_bounds__(32) void avg_final_kernel(
    const float* __restrict__ part,  // [AVG_BLOCKS, 32]
    float* __restrict__ out)         // 32 floats: soft[16] then hard[16]
{
  const int t = threadIdx.x;
  float s = 0.0f;
  for (int i = 0; i < AVG_BLOCKS; ++i) s += part[i * 32 + t];
  out[t] = s * (1.0f / (float)B_SZ);
}

// ---------------------------------------------------------------------------
extern "C" void kernel_launch(void* const* d_in, const int* in_sizes, int n_in,
                              void* d_out, int out_size, void* d_ws, size_t ws_size,
                              hipStream_t stream) {
  (void)in_sizes; (void)n_in; (void)out_size; (void)ws_size;
  const float* f    = (const float*)d_in[0];   // [B, D, E]
  const float* x    = (const float*)d_in[1];   // [B, DIN]
  const float* pw   = (const float*)d_in[2];   // [P, E, E]
  const float* gw   = (const float*)d_in[3];   // [E, DIN]
  const float* bias = (const float*)d_in[4];   // [E]

  float* out  = (float*)d_out;                 // y[B*D] ++ soft[16] ++ hard[16]
  float* g    = (float*)d_ws;                  // g_perm [B, E] = 512 KB scratch
  float* part = g + (size_t)B_SZ * E_SZ;       // [64, 32] partials = 8 KB scratch

  gate_kernel<<<B_SZ / 16, 32, 0, stream>>>(x, gw, pw, bias, g);
  avg_partial_kernel<<<AVG_BLOCKS, 32, 0, stream>>>(g, part);
  y_kernel<<<B_SZ, 256, 0, stream>>>(f, g, out);
  avg_final_kernel<<<1, 32, 0, stream>>>(part, out + (size_t)B_SZ * D_SZ);
}